// InertialEncoder_vinet_83296595739122
// MI455X (gfx1250) — compile-verified
//
#include <hip/hip_runtime.h>

#define HH 6      // hidden size
#define TT 1024   // sequence length
#define NG 24     // 4*H gates

typedef float v2f __attribute__((ext_vector_type(2)));
typedef float v8f __attribute__((ext_vector_type(8)));

__device__ __forceinline__ float d_tanh(float x) {
#if __has_builtin(__builtin_amdgcn_tanhf)
  return __builtin_amdgcn_tanhf(x);                 // v_tanh_f32
#elif __has_builtin(__builtin_amdgcn_tanh_f32)
  return __builtin_amdgcn_tanh_f32(x);
#else
  float e = __expf(2.0f * x);
  return 1.0f - 2.0f * __builtin_amdgcn_rcpf(e + 1.0f);
#endif
}

__device__ __forceinline__ v2f mkv2(float a, float b) {
  v2f v; v.x = a; v.y = b; return v;
}

__device__ __forceinline__ v8f wmma4(v2f a, v2f b, v8f c) {
  return __builtin_amdgcn_wmma_f32_16x16x4_f32(false, a, false, b, (short)0, c,
                                               false, false);
}

// Per-gate-row scale: sigmoid rows (i: 0-5, f: 6-11, o: 18-23) are pre-scaled
// by 0.5 so that sigmoid(g) = 0.5*tanh(g_scaled) + 0.5 with no runtime mul.
// tanh rows (g: 12-17) stay unscaled.
__device__ __forceinline__ float rowscale(int m) {
  return (m < 2 * HH || m >= 3 * HH) ? 0.5f : 1.0f;
}

// Element of combined weight W = [Wih | Whh]: row m (gate), col c (K in 0..11),
// with the sigmoid half-scale folded in per row.
__device__ __forceinline__ float welem(const float* __restrict__ wih,
                                       const float* __restrict__ whh,
                                       int m, int c) {
  if (m >= NG) return 0.0f;                         // padded rows of M-tile 1
  float w = (c < HH) ? wih[m * HH + c] : whh[m * HH + (c - HH)];
  return rowscale(m) * w;
}

// Per-batch-column LSTM state, replicated in both lane halves. Named scalar
// members only (no arrays) so everything stays in VGPRs.
struct HC {
  float h0, h1, h2, h3, h4, h5;
  float c0, c1, c2, c3, c4, c5;
};

// One LSTM cell step for 16 batch columns striped across a wave32.
// A{tile}{kchunk}: pre-scaled weights in WMMA A layout. cb0/cb1: pre-scaled
// bias in C/D layout. z0..z5 = input (Wih side), z6..z11 = h_prev (Whh side).
__device__ __forceinline__ HC lstm_cell(v2f A00, v2f A01, v2f A02,
                                        v2f A10, v2f A11, v2f A12,
                                        v8f cb0, v8f cb1, bool lo,
                                        float z0, float z1, float z2, float z3,
                                        float z4, float z5, float z6, float z7,
                                        float z8, float z9, float z10, float z11,
                                        HC s) {
  // B operand (K=4 x N=16): K=0,1 in lanes 0-15; K=2,3 in lanes 16-31.
  v2f b0 = mkv2(lo ? z0 : z2,  lo ? z1 : z3);
  v2f b1 = mkv2(lo ? z4 : z6,  lo ? z5 : z7);
  v2f b2 = mkv2(lo ? z8 : z10, lo ? z9 : z11);

  v8f acc0 = cb0;
  acc0 = wmma4(A00, b0, acc0);
  acc0 = wmma4(A01, b1, acc0);
  acc0 = wmma4(A02, b2, acc0);
  v8f acc1 = cb1;
  acc1 = wmma4(A10, b0, acc1);
  acc1 = wmma4(A11, b1, acc1);
  acc1 = wmma4(A12, b2, acc1);

  // tanh applied to the 16 LOCAL D values BEFORE the half-exchange: with the
  // half-scale folded into the weights, every gate activation starts with a
  // plain tanh, so the two lane halves split the tanh work instead of
  // duplicating it (16 + 6 trans per cell instead of 30). Padding rows of
  // tile 1 are exactly zero -> tanh(0)=0, harmless.
  float t00 = d_tanh(acc0[0]), t01 = d_tanh(acc0[1]);
  float t02 = d_tanh(acc0[2]), t03 = d_tanh(acc0[3]);
  float t04 = d_tanh(acc0[4]), t05 = d_tanh(acc0[5]);
  float t06 = d_tanh(acc0[6]), t07 = d_tanh(acc0[7]);
  float t10 = d_tanh(acc1[0]), t11 = d_tanh(acc1[1]);
  float t12 = d_tanh(acc1[2]), t13 = d_tanh(acc1[3]);
  float t14 = d_tanh(acc1[4]), t15 = d_tanh(acc1[5]);
  float t16 = d_tanh(acc1[6]), t17 = d_tanh(acc1[7]);

  // Gather tanh'd gates of this lane's batch column.
  // ga r -> tanh(gate r), gb r -> tanh(gate 8+r), gc r -> tanh(gate 16+r).
#define GATHER(ta, tb, ga, gb, gc)         \
  float ga, gb, gc;                        \
  {                                        \
    float o0 = __shfl_xor(ta, 16, 32);     \
    float o1 = __shfl_xor(tb, 16, 32);     \
    ga = lo ? ta : o0;                     \
    gb = lo ? o0 : ta;                     \
    gc = lo ? tb : o1;                     \
  }
  GATHER(t00, t10, ga0, gb0, gc0)
  GATHER(t01, t11, ga1, gb1, gc1)
  GATHER(t02, t12, ga2, gb2, gc2)
  GATHER(t03, t13, ga3, gb3, gc3)
  GATHER(t04, t14, ga4, gb4, gc4)
  GATHER(t05, t15, ga5, gb5, gc5)
  GATHER(t06, t16, ga6, gb6, gc6)
  GATHER(t07, t17, ga7, gb7, gc7)
#undef GATHER

  // PyTorch gate order i,f,g,o: i_j = gate j, f_j = gate 6+j,
  // g_j = gate 12+j, o_j = gate 18+j. Inputs here are tanh(gate).
#define CELL(iv, fv, gv, ov, hvar, cvar)                  \
  {                                                       \
    float ig = fmaf(0.5f, iv, 0.5f);                      \
    float fg = fmaf(0.5f, fv, 0.5f);                      \
    float og = fmaf(0.5f, ov, 0.5f);                      \
    float cn = fmaf(fg, cvar, ig * (gv));                 \
    cvar = cn;                                            \
    hvar = og * d_tanh(cn);                               \
  }
  CELL(ga0, ga6, gb4, gc2, s.h0, s.c0)
  CELL(ga1, ga7, gb5, gc3, s.h1, s.c1)
  CELL(ga2, gb0, gb6, gc4, s.h2, s.c2)
  CELL(ga3, gb1, gb7, gc5, s.h3, s.c3)
  CELL(ga4, gb2, gc0, gc6, s.h4, s.c4)
  CELL(ga5, gb3, gc1, gc7, s.h5, s.c5)
#undef CELL
  return s;
}

__global__ __launch_bounds__(32) void lstm2_wmma_kernel(
    const float* __restrict__ x,
    const float* __restrict__ wih0, const float* __restrict__ whh0,
    const float* __restrict__ bih0, const float* __restrict__ bhh0,
    const float* __restrict__ wih1, const float* __restrict__ whh1,
    const float* __restrict__ bih1, const float* __restrict__ bhh1,
    float* __restrict__ out, int B) {
  const int lane = threadIdx.x & 31;
  const bool lo = lane < 16;
  const int m0 = lane & 15;
  const int half = lo ? 0 : 1;
  const int nRaw = blockIdx.x * 16 + m0;        // batch column of this lane
  const int n = (nRaw < B) ? nRaw : (B - 1);
  const bool wr = lo && (nRaw < B);

  // ---- Weights in WMMA A layout (lanes 0-15: K=c,c+1; lanes 16-31: K=c+2,c+3)
  const int c0i = 2 * half, c1i = 4 + 2 * half, c2i = 8 + 2 * half;

  v2f A00l0 = mkv2(welem(wih0, whh0, m0, c0i),      welem(wih0, whh0, m0, c0i + 1));
  v2f A01l0 = mkv2(welem(wih0, whh0, m0, c1i),      welem(wih0, whh0, m0, c1i + 1));
  v2f A02l0 = mkv2(welem(wih0, whh0, m0, c2i),      welem(wih0, whh0, m0, c2i + 1));
  v2f A10l0 = mkv2(welem(wih0, whh0, 16 + m0, c0i), welem(wih0, whh0, 16 + m0, c0i + 1));
  v2f A11l0 = mkv2(welem(wih0, whh0, 16 + m0, c1i), welem(wih0, whh0, 16 + m0, c1i + 1));
  v2f A12l0 = mkv2(welem(wih0, whh0, 16 + m0, c2i), welem(wih0, whh0, 16 + m0, c2i + 1));

  v2f A00l1 = mkv2(welem(wih1, whh1, m0, c0i),      welem(wih1, whh1, m0, c0i + 1));
  v2f A01l1 = mkv2(welem(wih1, whh1, m0, c1i),      welem(wih1, whh1, m0, c1i + 1));
  v2f A02l1 = mkv2(welem(wih1, whh1, m0, c2i),      welem(wih1, whh1, m0, c2i + 1));
  v2f A10l1 = mkv2(welem(wih1, whh1, 16 + m0, c0i), welem(wih1, whh1, 16 + m0, c0i + 1));
  v2f A11l1 = mkv2(welem(wih1, whh1, 16 + m0, c1i), welem(wih1, whh1, 16 + m0, c1i + 1));
  v2f A12l1 = mkv2(welem(wih1, whh1, 16 + m0, c2i), welem(wih1, whh1, 16 + m0, c2i + 1));

  // ---- Bias folded into WMMA C operand (D layout: lo half M=r, hi half M=r+8),
  //      with the same per-row sigmoid half-scale.
  v8f cb0l0, cb1l0, cb0l1, cb1l1;
#pragma unroll
  for (int r = 0; r < 8; ++r) {
    const int r0 = lo ? r : (8 + r);
    cb0l0[r] = rowscale(r0) * (bih0[r0] + bhh0[r0]);
    cb1l0[r] = lo ? (rowscale(16 + r) * (bih0[16 + r] + bhh0[16 + r])) : 0.0f;
    cb0l1[r] = rowscale(r0) * (bih1[r0] + bhh1[r0]);
    cb1l1[r] = lo ? (rowscale(16 + r) * (bih1[16 + r] + bhh1[16 + r])) : 0.0f;
  }

  HC s1 = {0, 0, 0, 0, 0, 0, 0, 0, 0, 0, 0, 0};
  HC s2 = {0, 0, 0, 0, 0, 0, 0, 0, 0, 0, 0, 0};

  const float* __restrict__ xb = x + (size_t)n * TT * HH;
  float* __restrict__ ob = out + (size_t)n * TT * HH;

  // x(0) and x(1) (non-temporal: read-once stream).
  float xc0 = __builtin_nontemporal_load(xb + 0);
  float xc1 = __builtin_nontemporal_load(xb + 1);
  float xc2 = __builtin_nontemporal_load(xb + 2);
  float xc3 = __builtin_nontemporal_load(xb + 3);
  float xc4 = __builtin_nontemporal_load(xb + 4);
  float xc5 = __builtin_nontemporal_load(xb + 5);
  float xn0 = __builtin_nontemporal_load(xb + HH + 0);
  float xn1 = __builtin_nontemporal_load(xb + HH + 1);
  float xn2 = __builtin_nontemporal_load(xb + HH + 2);
  float xn3 = __builtin_nontemporal_load(xb + HH + 3);
  float xn4 = __builtin_nontemporal_load(xb + HH + 4);
  float xn5 = __builtin_nontemporal_load(xb + HH + 5);

  // ---- Prologue: layer-1 step 0.
  s1 = lstm_cell(A00l0, A01l0, A02l0, A10l0, A11l0, A12l0, cb0l0, cb1l0, lo,
                 xc0, xc1, xc2, xc3, xc4, xc5,
                 s1.h0, s1.h1, s1.h2, s1.h3, s1.h4, s1.h5, s1);

  // ---- Software-pipelined main loop: iteration t runs layer1(t) and
  // layer2(t-1). The two cells are data-independent (layer2 consumes the
  // PREVIOUS h1), so their WMMA/bpermute/tanh chains interleave and the
  // per-iteration critical path is ~one cell instead of two.
  for (int t = 1; t < TT; ++t) {
    // x(t+1) for the next iteration.
    const size_t off = (size_t)((t + 1 < TT) ? (t + 1) : t) * HH;
    float xf0 = __builtin_nontemporal_load(xb + off + 0);
    float xf1 = __builtin_nontemporal_load(xb + off + 1);
    float xf2 = __builtin_nontemporal_load(xb + off + 2);
    float xf3 = __builtin_nontemporal_load(xb + off + 3);
    float xf4 = __builtin_nontemporal_load(xb + off + 4);
    float xf5 = __builtin_nontemporal_load(xb + off + 5);

    // h1(t-1), consumed by layer2 below while s1 is being overwritten.
    const float p0 = s1.h0, p1 = s1.h1, p2 = s1.h2,
                p3 = s1.h3, p4 = s1.h4, p5 = s1.h5;

    // Layer 1, step t.
    s1 = lstm_cell(A00l0, A01l0, A02l0, A10l0, A11l0, A12l0, cb0l0, cb1l0, lo,
                   xn0, xn1, xn2, xn3, xn4, xn5,
                   p0, p1, p2, p3, p4, p5, s1);

    // Layer 2, step t-1 (independent of layer-1 step t above).
    s2 = lstm_cell(A00l1, A01l1, A02l1, A10l1, A11l1, A12l1, cb0l1, cb1l1, lo,
                   p0, p1, p2, p3, p4, p5,
                   s2.h0, s2.h1, s2.h2, s2.h3, s2.h4, s2.h5, s2);

    if (wr) {
      float* __restrict__ op = ob + (size_t)(t - 1) * HH;
      __builtin_nontemporal_store(s2.h0, op + 0);
      __builtin_nontemporal_store(s2.h1, op + 1);
      __builtin_nontemporal_store(s2.h2, op + 2);
      __builtin_nontemporal_store(s2.h3, op + 3);
      __builtin_nontemporal_store(s2.h4, op + 4);
      __builtin_nontemporal_store(s2.h5, op + 5);
    }

    xn0 = xf0; xn1 = xf1; xn2 = xf2; xn3 = xf3; xn4 = xf4; xn5 = xf5;
  }

  // ---- Epilogue: layer-2 step TT-1.
  s2 = lstm_cell(A00l1, A01l1, A02l1, A10l1, A11l1, A12l1, cb0l1, cb1l1, lo,
                 s1.h0, s1.h1, s1.h2, s1.h3, s1.h4, s1.h5,
                 s2.h0, s2.h1, s2.h2, s2.h3, s2.h4, s2.h5, s2);
  if (wr) {
    float* __restrict__ op = ob + (size_t)(TT - 1) * HH;
    __builtin_nontemporal_store(s2.h0, op + 0);
    __builtin_nontemporal_store(s2.h1, op + 1);
    __builtin_nontemporal_store(s2.h2, op + 2);
    __builtin_nontemporal_store(s2.h3, op + 3);
    __builtin_nontemporal_store(s2.h4, op + 4);
    __builtin_nontemporal_store(s2.h5, op + 5);
  }
}

extern "C" void kernel_launch(void* const* d_in, const int* in_sizes, int n_in,
                              void* d_out, int out_size, void* d_ws, size_t ws_size,
                              hipStream_t stream) {
  const float* x    = (const float*)d_in[0];
  const float* wih0 = (const float*)d_in[1];
  const float* whh0 = (const float*)d_in[2];
  const float* bih0 = (const float*)d_in[3];
  const float* bhh0 = (const float*)d_in[4];
  const float* wih1 = (const float*)d_in[5];
  const float* whh1 = (const float*)d_in[6];
  const float* bih1 = (const float*)d_in[7];
  const float* bhh1 = (const float*)d_in[8];
  float* out = (float*)d_out;

  const int B = in_sizes[0] / (TT * HH);        // 4096
  const int nblocks = (B + 15) / 16;            // 16 batch columns per wave

  hipLaunchKernelGGL(lstm2_wmma_kernel, dim3(nblocks), dim3(32), 0, stream,
                     x, wih0, whh0, bih0, bhh0, wih1, whh1, bih1, bhh1, out, B);
}